// GraphSAGE_70497593197182
// MI455X (gfx1250) — compile-verified
//
#include <hip/hip_runtime.h>

typedef __attribute__((ext_vector_type(2))) float v2f;
typedef __attribute__((ext_vector_type(8))) float v8f;

#define N_NODES 10000
#define IN_F    50
#define INP     64      // padded K for layer 1
#define HID     512
#define OUT_F   121
#define OUTP    128     // padded N for layer 2
#define N_EDGES 160000

// GEMM tiling
#define KC      64      // K chunk staged in LDS
#define NPANEL  128     // N panel per block (8 waves x 16)
#define SA_LD   66      // padded LDS row pitch for A rows (bank stride 2, conflict-free)
#define SBT_LD  68      // padded LDS row pitch for transposed B (bank stride 4, conflict-free)

// ---- workspace layout (float element offsets) ----
#define WS_DEG   0L
#define WS_AGG1  10240L
#define WS_AGG2  (WS_AGG1 + (long)N_NODES*INP)
#define WS_XP    (WS_AGG2 + (long)N_NODES*HID)
#define WS_W1L   (WS_XP  + (long)N_NODES*INP)
#define WS_W1R   (WS_W1L + (long)INP*HID)
#define WS_W2L   (WS_W1R + (long)INP*HID)
#define WS_W2R   (WS_W2L + (long)HID*OUTP)
#define WS_B2P   (WS_W2R + (long)HID*OUTP)
#define WS_H     (WS_B2P + OUTP)
#define WS_TOTAL (WS_H + (long)N_NODES*HID)   // ~11.73M floats ~= 47 MB

// ---------------- utility kernels ----------------
__global__ void k_zero(float* p, long n) {
  long i = (long)blockIdx.x * blockDim.x + threadIdx.x;
  long stride = (long)gridDim.x * blockDim.x;
  for (; i < n; i += stride) p[i] = 0.0f;
}

// copy rows x scols -> rows x dcols (zero column padding)
__global__ void k_copy2d(const float* __restrict__ src, float* __restrict__ dst,
                         int rows, int scols, int dcols) {
  int i = blockIdx.x * blockDim.x + threadIdx.x;
  int total = rows * dcols;
  if (i >= total) return;
  int r = i / dcols, c = i % dcols;
  dst[i] = (c < scols) ? src[r * scols + c] : 0.0f;
}

__global__ void k_degree(const int* __restrict__ dst_i, float* __restrict__ deg) {
  int e = blockIdx.x * blockDim.x + threadIdx.x;
  if (e < N_EDGES) atomicAdd(&deg[dst_i[e]], 1.0f);
}

// layer-1 scatter: 64 lanes per edge, feats 0..49
__global__ void k_scatter1(const float* __restrict__ x, const int* __restrict__ src_i,
                           const int* __restrict__ dst_i, float* __restrict__ agg1) {
  int idx = blockIdx.x * blockDim.x + threadIdx.x;
  int e = idx >> 6;
  int f = idx & 63;
  if (e >= N_EDGES || f >= IN_F) return;
  int s = src_i[e], d = dst_i[e];
  atomicAdd(&agg1[(long)d * INP + f], x[(long)s * IN_F + f]);
}

// layer-2 scatter: one block per edge over HID feats
__global__ void k_scatter2(const float* __restrict__ h, const int* __restrict__ src_i,
                           const int* __restrict__ dst_i, float* __restrict__ agg2) {
  int e = blockIdx.x;
  int s = src_i[e], d = dst_i[e];
  const float* hs = h + (long)s * HID;
  float* ad = agg2 + (long)d * HID;
  for (int f = threadIdx.x; f < HID; f += blockDim.x)
    atomicAdd(&ad[f], hs[f]);
}

// agg[n][c] *= 1/max(deg[n],1)
__global__ void k_meandiv(float* __restrict__ agg, const float* __restrict__ deg, int cols) {
  long i = (long)blockIdx.x * blockDim.x + threadIdx.x;
  long total = (long)N_NODES * cols;
  if (i >= total) return;
  int n = (int)(i / cols);
  float d = deg[n];
  agg[i] *= (d > 1.0f) ? (1.0f / d) : 1.0f;
}

// ---------------- LDS-staged fused two-term WMMA GEMM ----------------
// out[m, n] = sum_k A1[m,k]*B1[k,n] + sum_k A2[m,k]*B2[k,n] + bias[n]  (opt. ReLU)
// A1/A2: [M x K] row-major (lda == K). B1/B2: [K x ldb] row-major.
// Block = 256 threads (8 waves): one 16(M) x 128(N) output panel.
// B is staged TRANSPOSED in LDS (sBt[n][k]) so each WMMA reads one ds_load_b64
// for A and one for B.
__global__ void k_gemm_fused(const float* __restrict__ A1, const float* __restrict__ B1,
                             const float* __restrict__ A2, const float* __restrict__ B2,
                             const float* __restrict__ bias, float* __restrict__ out,
                             int K, int ldb, int ld_out, int out_cols, int relu) {
  __shared__ __align__(16) float sA[16 * SA_LD];        // A rows (row-major, pitch 66)
  __shared__ __align__(16) float sBt[NPANEL * SBT_LD];  // B transposed (n-major, pitch 68)

  int tid  = threadIdx.x;
  int wave = tid >> 5;
  int lane = tid & 31;
  int l16  = lane & 15;
  int half = lane >> 4;
  int m0   = blockIdx.y * 16;
  int n0p  = blockIdx.x * NPANEL;

  v8f c = {};
  for (int term = 0; term < 2; ++term) {
    const float* A = term ? A2 : A1;
    const float* B = term ? B2 : B1;
    for (int k0 = 0; k0 < K; k0 += KC) {
      __syncthreads();                       // previous chunk fully consumed
      // stage A chunk: 16 x KC (1024 elems, 4/thread), row-major
      for (int i = tid; i < 16 * KC; i += 256) {
        int r = i >> 6, cc = i & 63;         // KC == 64
        sA[r * SA_LD + cc] = A[(long)(m0 + r) * K + k0 + cc];
      }
      // stage B chunk: KC x NPANEL (8192 elems, 32/thread), TRANSPOSED into sBt[n][k]
      for (int i = tid; i < KC * NPANEL; i += 256) {
        int r = i >> 7, cc = i & 127;        // NPANEL == 128; coalesced global row read
        sBt[cc * SBT_LD + r] = B[(long)(k0 + r) * ldb + n0p + cc];
      }
      // prefetch next B chunk into cache while we compute this one
      if (k0 + KC < K && tid < KC)
        __builtin_prefetch(&B[(long)(k0 + KC + tid) * ldb + n0p], 0, 1);
      __syncthreads();                       // chunk staged

      const float* sa = &sA[l16 * SA_LD];                  // A row M = l16
      const float* sb = &sBt[(wave * 16 + l16) * SBT_LD];  // B col N = wave*16+l16
      #pragma unroll
      for (int kk = 0; kk < KC; kk += 4) {
        int ka = kk + 2 * half;              // even -> 8B-aligned v2f loads
        v2f a = *(const v2f*)(sa + ka);
        v2f b = *(const v2f*)(sb + ka);
        c = __builtin_amdgcn_wmma_f32_16x16x4_f32(false, a, false, b, (short)0, c, false, false);
      }
    }
  }

  int n = n0p + wave * 16 + l16;
  if (n < out_cols) {                        // divergence only after all WMMA done
    float bv = bias[n];
    #pragma unroll
    for (int r = 0; r < 8; ++r) {
      float v = c[r] + bv;
      if (relu) v = v > 0.0f ? v : 0.0f;
      out[(long)(m0 + r + 8 * half) * ld_out + n] = v;
    }
  }
}

// ---------------- launcher ----------------
extern "C" void kernel_launch(void* const* d_in, const int* in_sizes, int n_in,
                              void* d_out, int out_size, void* d_ws, size_t ws_size,
                              hipStream_t stream) {
  const float* x   = (const float*)d_in[0];
  const int*   ei  = (const int*)d_in[1];          // [2 x N_EDGES]: row0=src, row1=dst
  const float* W1l = (const float*)d_in[2];
  const float* W1r = (const float*)d_in[3];
  const float* b1  = (const float*)d_in[4];
  const float* W2l = (const float*)d_in[5];
  const float* W2r = (const float*)d_in[6];
  const float* b2  = (const float*)d_in[7];
  float* out = (float*)d_out;

  const int* src_i = ei;
  const int* dst_i = ei + N_EDGES;

  float* ws   = (float*)d_ws;
  float* deg  = ws + WS_DEG;
  float* agg1 = ws + WS_AGG1;
  float* agg2 = ws + WS_AGG2;
  float* xp   = ws + WS_XP;
  float* w1l  = ws + WS_W1L;
  float* w1r  = ws + WS_W1R;
  float* w2l  = ws + WS_W2L;
  float* w2r  = ws + WS_W2R;
  float* b2p  = ws + WS_B2P;
  float* h    = ws + WS_H;

  // 0) zero entire workspace (deg/agg must reset every call; pads stay zero)
  k_zero<<<4096, 256, 0, stream>>>(ws, WS_TOTAL);

  // 1) pad-copies
  k_copy2d<<<(N_NODES * INP + 255) / 256, 256, 0, stream>>>(x,   xp,  N_NODES, IN_F, INP);
  k_copy2d<<<(IN_F * HID + 255) / 256, 256, 0, stream>>>(W1l, w1l, IN_F, HID, HID); // rows 50..63 stay 0
  k_copy2d<<<(IN_F * HID + 255) / 256, 256, 0, stream>>>(W1r, w1r, IN_F, HID, HID);
  k_copy2d<<<(HID * OUTP + 255) / 256, 256, 0, stream>>>(W2l, w2l, HID, OUT_F, OUTP);
  k_copy2d<<<(HID * OUTP + 255) / 256, 256, 0, stream>>>(W2r, w2r, HID, OUT_F, OUTP);
  k_copy2d<<<(OUTP + 255) / 256, 256, 0, stream>>>(b2, b2p, 1, OUT_F, OUTP);

  // 2) degrees
  k_degree<<<(N_EDGES + 255) / 256, 256, 0, stream>>>(dst_i, deg);

  // 3) layer 1: scatter-sum, mean, fused GEMM (+bias, ReLU)
  k_scatter1<<<(N_EDGES * 64) / 256, 256, 0, stream>>>(x, src_i, dst_i, agg1);
  k_meandiv<<<(N_NODES * INP + 255) / 256, 256, 0, stream>>>(agg1, deg, INP);
  {
    dim3 grid(HID / NPANEL, N_NODES / 16);   // 4 x 625
    k_gemm_fused<<<grid, 256, 0, stream>>>(agg1, w1l, xp, w1r, b1, h,
                                           /*K=*/INP, /*ldb=*/HID,
                                           /*ld_out=*/HID, /*out_cols=*/HID, /*relu=*/1);
  }

  // 4) layer 2: scatter-sum, mean, fused GEMM (+bias)
  k_scatter2<<<N_EDGES, 256, 0, stream>>>(h, src_i, dst_i, agg2);
  k_meandiv<<<(N_NODES * HID + 255) / 256, 256, 0, stream>>>(agg2, deg, HID);
  {
    dim3 grid(OUTP / NPANEL, N_NODES / 16);  // 1 x 625
    k_gemm_fused<<<grid, 256, 0, stream>>>(agg2, w2l, h, w2r, b2p, out,
                                           /*K=*/HID, /*ldb=*/OUTP,
                                           /*ld_out=*/OUT_F, /*out_cols=*/OUT_F, /*relu=*/0);
  }
}